// LeafLayer_30777735643843
// MI455X (gfx1250) — compile-verified
//
#include <hip/hip_runtime.h>

// Problem constants (from reference)
#define BATCH 4
#define NDIM  512
#define MDIM  128
#define HDIM  128
#define BN    (BATCH * NDIM)   // 2048 total rows

typedef __attribute__((ext_vector_type(2))) float v2f;
typedef __attribute__((ext_vector_type(8))) float v8f;

// ---------------------------------------------------------------------------
// Kernel 1: q = h@Wq^T + bq ; v = (h@Wv^T + bv)*10
// Grid: BN/16 = 128 blocks, 256 threads (8 waves). Wave w -> output cols
// [16w, 16w+16). One A-fragment load feeds both accumulators.
// ---------------------------------------------------------------------------
__global__ __launch_bounds__(256) void qv_kernel(
    const float* __restrict__ h,                 // [BN, MDIM]
    const float* __restrict__ Wv, const float* __restrict__ bv,
    const float* __restrict__ Wq, const float* __restrict__ bq,
    float* __restrict__ vout,                    // [BN, HDIM]
    float* __restrict__ qout)                    // [BN, HDIM]
{
  const int lane = threadIdx.x & 31;
  const int wave = threadIdx.x >> 5;
  const int lm   = lane & 15;
  const int half = lane >> 4;
  const int row0 = blockIdx.x * 16;
  const int col0 = wave * 16;
  const int n    = col0 + lm;

  v8f accv = {};
  v8f accq = {};
  const float* arow = h + (size_t)(row0 + lm) * MDIM;
  const float* wvr  = Wv + (size_t)n * MDIM;    // Wv[n, :]  (B[k][n] = Wv[n][k])
  const float* wqr  = Wq + (size_t)n * MDIM;

  for (int k = 0; k < MDIM; k += 4) {
    const int kk = k + 2 * half;
    v2f a;  a.x  = arow[kk];  a.y  = arow[kk + 1];
    v2f bv_; bv_.x = wvr[kk]; bv_.y = wvr[kk + 1];
    v2f bq_; bq_.x = wqr[kk]; bq_.y = wqr[kk + 1];
    accv = __builtin_amdgcn_wmma_f32_16x16x4_f32(false, a, false, bv_,
                                                 (short)0, accv, false, false);
    accq = __builtin_amdgcn_wmma_f32_16x16x4_f32(false, a, false, bq_,
                                                 (short)0, accq, false, false);
  }

  const float biasv = bv[n];
  const float biasq = bq[n];
#pragma unroll
  for (int r = 0; r < 8; ++r) {
    const int row = row0 + r + 8 * half;
    vout[(size_t)row * HDIM + n] = (accv[r] + biasv) * 10.0f;
    qout[(size_t)row * HDIM + n] =  accq[r] + biasq;
  }
}

// ---------------------------------------------------------------------------
// Kernel 2: qw[row] = q[row,:].wd ; qb[row] = q[row,:].bd   (tiny, mem-bound)
// ---------------------------------------------------------------------------
__global__ __launch_bounds__(256) void qwqb_kernel(
    const float* __restrict__ q,
    const float* __restrict__ wd, const float* __restrict__ bd,
    float* __restrict__ qw, float* __restrict__ qb)
{
  const int row = blockIdx.x * blockDim.x + threadIdx.x;
  if (row >= BN) return;
  const float* qr = q + (size_t)row * HDIM;
  float sw = 0.f, sb = 0.f;
#pragma unroll 4
  for (int k = 0; k < HDIM; ++k) {
    const float x = qr[k];
    sw += x * wd[k];
    sb += x * bd[k];
  }
  qw[row] = sw;
  qb[row] = sb;
}

// ---------------------------------------------------------------------------
// Kernel 3: att[row, :] = softmax(d[row, :] * qw[row] + qb[row])
// One block (256 threads) per row; each thread owns 2 of the 512 columns.
// wave32 shuffle reduce + LDS cross-wave reduce.
// ---------------------------------------------------------------------------
__global__ __launch_bounds__(256) void softmax_kernel(
    const float* __restrict__ d,
    const float* __restrict__ qw, const float* __restrict__ qb,
    float* __restrict__ att)
{
  const int row = blockIdx.x;                      // 0..BN-1
  const int b   = row / NDIM;
  const int i   = row % NDIM;
  const float* drow = d + ((size_t)b * NDIM + i) * NDIM;
  float*       arow = att + (size_t)row * NDIM;

  const float sqw = qw[row];
  const float sqb = qb[row];

  const int t = threadIdx.x;
  const float x0 = fmaf(drow[t],       sqw, sqb);
  const float x1 = fmaf(drow[t + 256], sqw, sqb);

  __shared__ float redm[8];
  __shared__ float reds[8];

  // --- max reduce ---
  float m = fmaxf(x0, x1);
#pragma unroll
  for (int off = 16; off > 0; off >>= 1)
    m = fmaxf(m, __shfl_xor(m, off, 32));
  if ((t & 31) == 0) redm[t >> 5] = m;
  __syncthreads();
  float gm = redm[0];
#pragma unroll
  for (int w = 1; w < 8; ++w) gm = fmaxf(gm, redm[w]);

  // --- exp + sum reduce ---
  const float e0 = __expf(x0 - gm);
  const float e1 = __expf(x1 - gm);
  float s = e0 + e1;
#pragma unroll
  for (int off = 16; off > 0; off >>= 1)
    s += __shfl_xor(s, off, 32);
  if ((t & 31) == 0) reds[t >> 5] = s;
  __syncthreads();
  float gs = 0.f;
#pragma unroll
  for (int w = 0; w < 8; ++w) gs += reds[w];
  const float inv = 1.0f / gs;

  arow[t]       = e0 * inv;
  arow[t + 256] = e1 * inv;
}

// ---------------------------------------------------------------------------
// Kernel 4: out = att @ v  (batched GEMM, K = 512, f32 WMMA)
// Grid: BN/16 = 128 blocks x 8 waves. B-matrix is row-major [N x H] (v).
// ---------------------------------------------------------------------------
__global__ __launch_bounds__(256) void attv_kernel(
    const float* __restrict__ att,               // [BATCH, NDIM, NDIM]
    const float* __restrict__ v,                 // [BN, HDIM]
    float* __restrict__ out)                     // [BN, HDIM]
{
  const int lane = threadIdx.x & 31;
  const int wave = threadIdx.x >> 5;
  const int lm   = lane & 15;
  const int half = lane >> 4;
  const int row0 = blockIdx.x * 16;              // global row over BN
  const int b    = row0 / NDIM;
  const int i0   = row0 % NDIM;
  const int col0 = wave * 16;
  const int n    = col0 + lm;

  const float* A  = att + (size_t)b * NDIM * NDIM;
  const float* Bm = v   + (size_t)b * NDIM * HDIM;
  const float* arow = A + (size_t)(i0 + lm) * NDIM;

  v8f acc = {};
  for (int k = 0; k < NDIM; k += 4) {
    const int kk = k + 2 * half;
    v2f a;  a.x = arow[kk];     a.y = arow[kk + 1];
    v2f bf; bf.x = Bm[(size_t)kk * HDIM + n];
            bf.y = Bm[(size_t)(kk + 1) * HDIM + n];
    acc = __builtin_amdgcn_wmma_f32_16x16x4_f32(false, a, false, bf,
                                                (short)0, acc, false, false);
  }

#pragma unroll
  for (int r = 0; r < 8; ++r)
    out[(size_t)(row0 + r + 8 * half) * HDIM + n] = acc[r];
}

// ---------------------------------------------------------------------------
// Kernel 5/6: Y = act(X @ W^T + bias); W stored [OUT x K] row-major (K=OUT=128)
// ---------------------------------------------------------------------------
template <bool RELU>
__global__ __launch_bounds__(256) void fc_kernel(
    const float* __restrict__ X,                 // [BN, 128]
    const float* __restrict__ W,                 // [128, 128]
    const float* __restrict__ bias,              // [128]
    float* __restrict__ Y)                       // [BN, 128]
{
  const int lane = threadIdx.x & 31;
  const int wave = threadIdx.x >> 5;
  const int lm   = lane & 15;
  const int half = lane >> 4;
  const int row0 = blockIdx.x * 16;
  const int col0 = wave * 16;
  const int n    = col0 + lm;

  const float* arow = X + (size_t)(row0 + lm) * 128;
  const float* wr   = W + (size_t)n * 128;

  v8f acc = {};
  for (int k = 0; k < 128; k += 4) {
    const int kk = k + 2 * half;
    v2f a;  a.x = arow[kk]; a.y = arow[kk + 1];
    v2f bf; bf.x = wr[kk];  bf.y = wr[kk + 1];
    acc = __builtin_amdgcn_wmma_f32_16x16x4_f32(false, a, false, bf,
                                                (short)0, acc, false, false);
  }

  const float bn = bias[n];
#pragma unroll
  for (int r = 0; r < 8; ++r) {
    float y = acc[r] + bn;
    if (RELU) y = fmaxf(y, 0.0f);
    Y[(size_t)(row0 + r + 8 * half) * 128 + n] = y;
  }
}

// ---------------------------------------------------------------------------
// Host-side launcher
// ---------------------------------------------------------------------------
extern "C" void kernel_launch(void* const* d_in, const int* in_sizes, int n_in,
                              void* d_out, int out_size, void* d_ws, size_t ws_size,
                              hipStream_t stream) {
  const float* d  = (const float*)d_in[0];   // [B, N, N]
  const float* h  = (const float*)d_in[1];   // [B, N, M]
  const float* Wv = (const float*)d_in[2];
  const float* bv = (const float*)d_in[3];
  const float* Wq = (const float*)d_in[4];
  const float* bq = (const float*)d_in[5];
  const float* wd = (const float*)d_in[6];
  const float* bd = (const float*)d_in[7];
  const float* W1 = (const float*)d_in[8];
  const float* b1 = (const float*)d_in[9];
  const float* W2 = (const float*)d_in[10];
  const float* b2 = (const float*)d_in[11];
  float* y = (float*)d_out;                  // [B, N, M]

  // Workspace layout (floats)
  float* ws   = (float*)d_ws;
  float* q    = ws;                            // BN*H   = 262144
  float* v    = q    + (size_t)BN * HDIM;      // BN*H
  float* qw   = v    + (size_t)BN * HDIM;      // BN
  float* qb   = qw   + BN;                     // BN
  float* att  = qb   + BN;                     // B*N*N  = 1048576
  float* out  = att  + (size_t)BATCH * NDIM * NDIM;  // BN*H
  float* tmid = out  + (size_t)BN * HDIM;      // BN*H

  qv_kernel<<<BN / 16, 256, 0, stream>>>(h, Wv, bv, Wq, bq, v, q);
  qwqb_kernel<<<BN / 256, 256, 0, stream>>>(q, wd, bd, qw, qb);
  softmax_kernel<<<BN, 256, 0, stream>>>(d, qw, qb, att);
  attv_kernel<<<BN / 16, 256, 0, stream>>>(att, v, out);
  fc_kernel<true><<<BN / 16, 256, 0, stream>>>(out, W1, b1, tmid);
  fc_kernel<false><<<BN / 16, 256, 0, stream>>>(tmid, W2, b2, y);
}